// WNetLoss_61349312856563
// MI455X (gfx1250) — compile-verified
//
#include <hip/hip_runtime.h>
#include <math.h>

// ---------------- problem constants (fixed by reference) ----------------
#define B_  8
#define C_  3
#define K_  16
#define H_  512
#define W_  512
#define HW  (H_*W_)
#define R_  4
#define TS  16
#define HT  (TS + 2*R_)      // 24 (haloed tile)

typedef float v2f __attribute__((ext_vector_type(2)));
typedef float v8f __attribute__((ext_vector_type(8)));

#if __has_builtin(__builtin_amdgcn_global_load_async_to_lds_b32) && \
    __has_builtin(__builtin_amdgcn_s_wait_asynccnt)
#define HAS_ASYNC_LDS 1
typedef __attribute__((address_space(1))) int g_i32;   // global (AS1)
typedef __attribute__((address_space(3))) int l_i32;   // LDS    (AS3)
#endif

// ---------------- workspace layout (floats) ----------------
#define WS_SUMCP   0                        // [B_*K_]        sum(cp)
#define WS_SUMICP  (WS_SUMCP + B_*K_)       // [B_*K_*3]      sum(in_c*cp)
#define WS_NUM     (WS_SUMICP + B_*K_*3)    // [K_]
#define WS_DEN     (WS_NUM + K_)            // [K_]
#define WS_MSE     (WS_DEN + K_)            // [1]
#define WS_SMOOTH  (WS_MSE + 1)             // [1]
#define WS_TOTAL   (WS_SMOOTH + 1)

__device__ __forceinline__ float waveRed(float v) {
    #pragma unroll
    for (int off = 16; off > 0; off >>= 1) v += __shfl_down(v, off);
    return v;
}

// ---------------------------------------------------------------------------
// Pass 1: per-(b,k) sums for cm  +  grey-opening smoothness term.
// ---------------------------------------------------------------------------
__global__ __launch_bounds__(256)
void stats_smooth_kernel(const float* __restrict__ in,
                         const float* __restrict__ msk,
                         float* __restrict__ ws) {
    const int b   = blockIdx.x >> 10;                 // 1024 blocks / image
    const int pix = ((blockIdx.x & 1023) << 8) + threadIdx.x;
    const int y   = pix >> 9;
    const int x   = pix & (W_ - 1);
    const int lane = threadIdx.x & 31;

    const float* inb = in + (size_t)b * C_ * HW;
    const float i0 = inb[pix];
    const float i1 = inb[HW + pix];
    const float i2 = inb[2 * HW + pix];

    __shared__ float sAcc[K_ * 4 + 1];
    if (threadIdx.x < K_ * 4 + 1) sAcc[threadIdx.x] = 0.f;
    __syncthreads();

    float smooth = 0.f;
    for (int k = 0; k < K_; ++k) {
        const float* mk = msk + (size_t)(b * K_ + k) * HW;
        const float cp  = mk[pix];

        // grey_opening_2x2 with symmetric (edge) padding
        auto eat = [&](int qy, int qx) -> float {
            qy = min(qy, H_ - 1); qx = min(qx, W_ - 1);
            const int ay = max(qy - 1, 0), ax = max(qx - 1, 0);
            return fminf(fminf(mk[ay * W_ + ax], mk[ay * W_ + qx]),
                         fminf(mk[qy * W_ + ax], mk[qy * W_ + qx]));
        };
        const float o = fmaxf(fmaxf(eat(y, x), eat(y, x + 1)),
                              fmaxf(eat(y + 1, x), eat(y + 1, x + 1)));
        const float dlt = cp - o;
        smooth += dlt * dlt;

        float v0 = waveRed(cp);
        float v1 = waveRed(i0 * cp);
        float v2 = waveRed(i1 * cp);
        float v3 = waveRed(i2 * cp);
        if (lane == 0) {
            atomicAdd(&sAcc[k * 4 + 0], v0);
            atomicAdd(&sAcc[k * 4 + 1], v1);
            atomicAdd(&sAcc[k * 4 + 2], v2);
            atomicAdd(&sAcc[k * 4 + 3], v3);
        }
    }
    smooth = waveRed(smooth);
    if (lane == 0) atomicAdd(&sAcc[K_ * 4], smooth);
    __syncthreads();

    if (threadIdx.x < K_ * 4) {
        const int k = threadIdx.x >> 2, c = threadIdx.x & 3;
        const float v = sAcc[threadIdx.x];
        if (c == 0) atomicAdd(&ws[WS_SUMCP + b * K_ + k], v);
        else        atomicAdd(&ws[WS_SUMICP + (b * K_ + k) * 3 + (c - 1)], v);
    } else if (threadIdx.x == K_ * 4) {
        atomicAdd(&ws[WS_SMOOTH], sAcc[K_ * 4]);
    }
}

// ---------------------------------------------------------------------------
// Pass 2: fused ncut kernel. One wave per 16x16 output tile.
// Separable 9-tap Gaussian conv via V_WMMA_F32_16X16X4_F32 banded matmuls.
// Class-(k+1) cp tile is fetched with GLOBAL_LOAD_ASYNC_TO_LDS_B32 into a
// double-buffered LDS tile while class k computes (ASYNCcnt tracked).
// ---------------------------------------------------------------------------
#ifdef HAS_ASYNC_LDS
__device__ __forceinline__ void async_tile_load(const float* __restrict__ gbase,
                                                float (*dst)[HT],
                                                int y0, int x0, int lane) {
    #pragma unroll
    for (int it = 0; it < (HT * HT) / 32; ++it) {       // 18 iterations, exact
        const int idx = it * 32 + lane;
        const int r = idx / HT, c = idx % HT;
        const int gy = y0 - R_ + r, gx = x0 - R_ + c;
        if (gy >= 0 && gy < H_ && gx >= 0 && gx < W_) {
            __builtin_amdgcn_global_load_async_to_lds_b32(
                (g_i32*)(gbase + gy * W_ + gx),
                (l_i32*)&dst[r][c],
                0, 0);
        }
    }
}
#endif

__global__ __launch_bounds__(32)
void ncut_kernel(const float* __restrict__ in,
                 const float* __restrict__ msk,
                 float* __restrict__ ws) {
    __shared__ float sIn[3][HT][HT];   // input tile, zero-padded outside image
    __shared__ float sCp[2][HT][HT];   // double-buffered class-prob tile (raw, 0-padded)
    __shared__ float sW [HT][HT];      // affinity tile
    __shared__ float sG [HT][TS];      // banded Gaussian  G[j][c] = g[j-c]
    __shared__ float sTw [HT][TS];     // horizontal conv of w
    __shared__ float sTcw[HT][TS];     // horizontal conv of cp*w

    const int lane = threadIdx.x;      // 0..31, wave32
    const int hlf  = lane >> 4;
    const int m    = lane & 15;
    const int x0   = blockIdx.x * TS;
    const int y0   = blockIdx.y * TS;
    const int b    = blockIdx.z;
    const float* inb = in + (size_t)b * C_ * HW;

#ifdef HAS_ASYNC_LDS
    // zero the (k-invariant) out-of-image halo cells once, then kick off k=0
    for (int idx = lane; idx < HT * HT; idx += 32) {
        const int r = idx / HT, c = idx % HT;
        const int gy = y0 - R_ + r, gx = x0 - R_ + c;
        if (!((gy >= 0) & (gy < H_) & (gx >= 0) & (gx < W_))) {
            sCp[0][r][c] = 0.f;
            sCp[1][r][c] = 0.f;
        }
    }
    async_tile_load(msk + (size_t)(b * K_) * HW, sCp[0], y0, x0, lane);
#endif

    // ---- load 3-channel input tile (zero pad outside image: conv uses 0-pad)
    for (int idx = lane; idx < HT * HT; idx += 32) {
        const int r = idx / HT, c = idx % HT;
        const int gy = y0 - R_ + r, gx = x0 - R_ + c;
        const bool ok = (gy >= 0) & (gy < H_) & (gx >= 0) & (gx < W_);
        const int p = gy * W_ + gx;
        sIn[0][r][c] = ok ? inb[p]          : 0.f;
        sIn[1][r][c] = ok ? inb[HW + p]     : 0.f;
        sIn[2][r][c] = ok ? inb[2 * HW + p] : 0.f;
    }
    // ---- banded separable Gaussian: g(t) = exp(-(t-4)^2 / (2*sigma1^2)), sigma1=5
    for (int idx = lane; idx < HT * TS; idx += 32) {
        const int j = idx / TS, c = idx % TS, d = j - c;
        sG[j][c] = (d >= 0 && d <= 8)
                       ? __expf(-(float)((d - 4) * (d - 4)) * 0.02f)
                       : 0.f;
    }
    __syncthreads();

    for (int k = 0; k < K_; ++k) {
        const int buf = k & 1;
        if (k + 2 < K_) {   // L2 warm-up two classes ahead (global_prefetch_b8)
            __builtin_prefetch(msk + (size_t)(b * K_ + k + 2) * HW
                                   + (size_t)max(y0 - R_, 0) * W_ + x0, 0, 1);
        }

#ifdef HAS_ASYNC_LDS
        __builtin_amdgcn_s_wait_asynccnt(0);     // class-k tile resident in LDS
        if (k + 1 < K_)                          // overlap k+1 fetch with k compute
            async_tile_load(msk + (size_t)(b * K_ + k + 1) * HW,
                            sCp[buf ^ 1], y0, x0, lane);
#else
        {   // synchronous fallback tile load
            const float* mk = msk + (size_t)(b * K_ + k) * HW;
            for (int idx = lane; idx < HT * HT; idx += 32) {
                const int r = idx / HT, c = idx % HT;
                const int gy = y0 - R_ + r, gx = x0 - R_ + c;
                const bool ok = (gy >= 0) & (gy < H_) & (gx >= 0) & (gx < W_);
                sCp[buf][r][c] = ok ? mk[gy * W_ + gx] : 0.f;
            }
        }
#endif
        // cm_{b,k,c} from pass-1 sums: mean(in*cp) / (mean(cp)+1e-5)
        const float sc  = ws[WS_SUMCP + b * K_ + k];
        const float inv = 1.0f / (sc + 1e-5f * (float)HW);
        const float cm0 = ws[WS_SUMICP + (b * K_ + k) * 3 + 0] * inv;
        const float cm1 = ws[WS_SUMICP + (b * K_ + k) * 3 + 1] * inv;
        const float cm2 = ws[WS_SUMICP + (b * K_ + k) * 3 + 2] * inv;

        // ---- affinity w = exp(-diff^2), zero outside image
        for (int idx = lane; idx < HT * HT; idx += 32) {
            const int r = idx / HT, c = idx % HT;
            const int gy = y0 - R_ + r, gx = x0 - R_ + c;
            const bool ok = (gy >= 0) & (gy < H_) & (gx >= 0) & (gx < W_);
            const float d0 = sIn[0][r][c] - cm0;
            const float d1 = sIn[1][r][c] - cm1;
            const float d2 = sIn[2][r][c] - cm2;
            const float diff = d0 * d0 + d1 * d1 + d2 * d2;
            sW[r][c] = ok ? __expf(-diff * diff) : 0.f;   // sigma2 = 1
        }
        __syncthreads();

        // ---- horizontal pass: tmp[r][c] = sum_t src[r][c+t]*g[t]
        #pragma unroll
        for (int rg = 0; rg < 2; ++rg) {
            const int sr0 = rg * 8;
            v8f aw = {}, acw = {};
            #pragma unroll
            for (int kk = 0; kk < 6; ++kk) {
                const int j0 = kk * 4 + 2 * hlf;
                v2f bg; bg.x = sG[j0][m]; bg.y = sG[j0 + 1][m];
                const int row = sr0 + m;
                v2f a1; a1.x = sW[row][j0]; a1.y = sW[row][j0 + 1];
                aw = __builtin_amdgcn_wmma_f32_16x16x4_f32(
                        false, a1, false, bg, (short)0, aw, false, false);
                v2f a2; a2.x = a1.x * sCp[buf][row][j0];
                        a2.y = a1.y * sCp[buf][row][j0 + 1];
                acw = __builtin_amdgcn_wmma_f32_16x16x4_f32(
                        false, a2, false, bg, (short)0, acw, false, false);
            }
            #pragma unroll
            for (int i = 0; i < 8; ++i) {
                sTw [sr0 + i + 8 * hlf][m] = aw[i];
                sTcw[sr0 + i + 8 * hlf][m] = acw[i];
            }
        }
        __syncthreads();

        // ---- vertical pass: out[r][c] = sum_t tmp[r+t][c]*g[t]  (A = band, B = tmp)
        v8f vw = {}, vcw = {};
        #pragma unroll
        for (int kk = 0; kk < 6; ++kk) {
            const int j0 = kk * 4 + 2 * hlf;
            v2f ag; ag.x = sG[j0][m]; ag.y = sG[j0 + 1][m];
            v2f bw; bw.x = sTw[j0][m]; bw.y = sTw[j0 + 1][m];
            vw = __builtin_amdgcn_wmma_f32_16x16x4_f32(
                    false, ag, false, bw, (short)0, vw, false, false);
            v2f bc; bc.x = sTcw[j0][m]; bc.y = sTcw[j0 + 1][m];
            vcw = __builtin_amdgcn_wmma_f32_16x16x4_f32(
                    false, ag, false, bc, (short)0, vcw, false, false);
        }

        // ---- num/den partial dot with cp (interior 16x16)
        float numv = 0.f, denv = 0.f;
        #pragma unroll
        for (int i = 0; i < 8; ++i) {
            const float cp = sCp[buf][R_ + i + 8 * hlf][R_ + m];
            numv += cp * vcw[i];
            denv += cp * vw[i];
        }
        numv = waveRed(numv);
        denv = waveRed(denv);
        if (lane == 0) {
            atomicAdd(&ws[WS_NUM + k], numv);
            atomicAdd(&ws[WS_DEN + k], denv);
        }
        __syncthreads();   // before next class overwrites sW/sTw/sTcw
    }
}

// ---------------------------------------------------------------------------
// MSE reduction over output/input
// ---------------------------------------------------------------------------
__global__ __launch_bounds__(256)
void mse_kernel(const float* __restrict__ o,
                const float* __restrict__ in,
                float* __restrict__ ws) {
    const size_t i = (size_t)blockIdx.x * 256 + threadIdx.x;
    const float d = o[i] - in[i];
    float v = waveRed(d * d);
    if ((threadIdx.x & 31) == 0) atomicAdd(&ws[WS_MSE], v);
}

// ---------------------------------------------------------------------------
// Finalize scalar loss
// ---------------------------------------------------------------------------
__global__ void final_kernel(const float* __restrict__ ws, float* __restrict__ out) {
    float s = 0.f;
    for (int k = 0; k < K_; ++k)
        s += fabsf(ws[WS_NUM + k] / (ws[WS_DEN + k] + 1e-6f));
    const float ncut   = 0.001f * ((float)K_ - s);
    const float mse    = 1.0f * ws[WS_MSE] / (float)(B_ * C_ * HW);
    const float smooth = 0.1f * ws[WS_SMOOTH] / (float)(B_ * K_ * HW);
    out[0] = ncut + mse + smooth;
}

// ---------------------------------------------------------------------------
extern "C" void kernel_launch(void* const* d_in, const int* in_sizes, int n_in,
                              void* d_out, int out_size, void* d_ws, size_t ws_size,
                              hipStream_t stream) {
    const float* outp = (const float*)d_in[0];   // 'output'
    // d_in[1] = 'target' (unused by reference)
    const float* inp  = (const float*)d_in[2];   // 'input'
    const float* mask = (const float*)d_in[3];   // 'mask'
    float* ws   = (float*)d_ws;
    float* outf = (float*)d_out;

    (void)hipMemsetAsync(d_ws, 0, WS_TOTAL * sizeof(float), stream);

    stats_smooth_kernel<<<dim3(B_ * (HW / 256)), 256, 0, stream>>>(inp, mask, ws);
    mse_kernel<<<dim3((B_ * C_ * HW) / 256), 256, 0, stream>>>(outp, inp, ws);
    ncut_kernel<<<dim3(W_ / TS, H_ / TS, B_), 32, 0, stream>>>(inp, mask, ws);
    final_kernel<<<1, 1, 0, stream>>>(ws, outf);
}